// FCOSOutputs_79293686219124
// MI455X (gfx1250) — compile-verified
//
#include <hip/hip_runtime.h>
#include <stdint.h>

#define INF_C   100000000.0f
#define BG_ID_C 100000
#define RADIUS_C 1.5f
#define MAXG 64
#define BLOCK 256

// LDS layout per GT: [x1, y1, x2, y2, cx, cy, area, pad]  (8 floats, so float4-pair reads)
__global__ __launch_bounds__(BLOCK) void fcos_assign_kernel(
    const float* __restrict__ locations,       // [L,2]
    const float* __restrict__ size_ranges,     // [L,2]
    const float* __restrict__ stride_per_loc,  // [L]
    const float* __restrict__ gt_boxes,        // [B,G,4]
    const int*   __restrict__ gt_classes,      // [B,G]
    float* __restrict__ out,                   // [B*L*5] ++ [B*L] ++ [B*L]
    int L, int G, int B)
{
    __shared__ float sbox[MAXG * 8];
    __shared__ int   scls[MAXG];

    const int tid = threadIdx.x;
    const int b   = blockIdx.y;
    const int loc = blockIdx.x * BLOCK + tid;

    // ---- Stage this batch's GT boxes + classes into LDS via CDNA5 async loads ----
    if (tid < G) {
        uint32_t lds_box = (uint32_t)(uintptr_t)(&sbox[tid * 8]);
        uint64_t ga_box  = (uint64_t)(uintptr_t)(gt_boxes + ((size_t)b * G + tid) * 4);
        asm volatile("global_load_async_to_lds_b128 %0, %1, off"
                     :: "v"(lds_box), "v"(ga_box) : "memory");
        uint32_t lds_cls = (uint32_t)(uintptr_t)(&scls[tid]);
        uint64_t ga_cls  = (uint64_t)(uintptr_t)(gt_classes + (size_t)b * G + tid);
        asm volatile("global_load_async_to_lds_b32 %0, %1, off"
                     :: "v"(lds_cls), "v"(ga_cls) : "memory");
    }
    asm volatile("s_wait_asynccnt 0x0" ::: "memory");
    __syncthreads();

    // Precompute per-GT center + area once per block (broadcast data).
    if (tid < G) {
        float x1 = sbox[tid * 8 + 0], y1 = sbox[tid * 8 + 1];
        float x2 = sbox[tid * 8 + 2], y2 = sbox[tid * 8 + 3];
        sbox[tid * 8 + 4] = (x1 + x2) * 0.5f;
        sbox[tid * 8 + 5] = (y1 + y2) * 0.5f;
        sbox[tid * 8 + 6] = (x2 - x1) * (y2 - y1);
        sbox[tid * 8 + 7] = 0.0f;
    }
    __syncthreads();

    if (loc >= L) return;

    // Per-location geometry (coalesced b64 loads).
    const float2 xy = *(const float2*)(locations + (size_t)loc * 2);
    const float2 sr = *(const float2*)(size_ranges + (size_t)loc * 2);
    const float  x = xy.x, y = xy.y, lo = sr.x, hi = sr.y;
    const float  stride = stride_per_loc[loc];
    const float  rs = stride * RADIUS_C;

    const float4* sb4 = reinterpret_cast<const float4*>(sbox);

    float best = INF_C;
    int   bidx = 0;
#pragma unroll 4
    for (int g = 0; g < G; ++g) {
        const float4 bx = sb4[g * 2 + 0];   // x1,y1,x2,y2  (broadcast ds_load_b128)
        const float4 cc = sb4[g * 2 + 1];   // cx,cy,area,-
        const float l = x - bx.x;
        const float t = y - bx.y;
        const float r = bx.z - x;
        const float d = bx.w - y;
        const float mr   = fmaxf(fmaxf(l, t), fmaxf(r, d));
        const float xmin = fmaxf(cc.x - rs, bx.x);
        const float xmax = fminf(cc.x + rs, bx.z);
        const float ymin = fmaxf(cc.y - rs, bx.y);
        const float ymax = fminf(cc.y + rs, bx.w);
        const float ins  = fminf(fminf(x - xmin, xmax - x), fminf(y - ymin, ymax - y));
        const bool ok = (ins > 0.0f) && (mr >= lo) && (mr <= hi);
        const float a = ok ? cc.z : INF_C;
        if (a < best) { best = a; bidx = g; }   // strict < => first min wins (jnp.argmin)
    }

    // Gather the selected box (divergent LDS read is fine) and finish.
    const float4 bb = sb4[bidx * 2];
    const float l = x - bb.x;
    const float t = y - bb.y;
    const float r = bb.z - x;
    const float d = bb.w - y;
    const float minlr = fminf(l, r), maxlr = fmaxf(l, r);
    const float mintb = fminf(t, d), maxtb = fmaxf(t, d);
    const float ratio = (minlr / maxlr) * (mintb / maxtb);
    const float ctr = (ratio > 0.0f) ? __builtin_sqrtf(ratio) : 0.0f;

    const size_t base = (size_t)b * L + loc;
    float* fo = out + base * 5;
    fo[0] = l / stride;
    fo[1] = t / stride;
    fo[2] = r / stride;
    fo[3] = d / stride;
    fo[4] = ctr;

    const int label = (best >= INF_C) ? BG_ID_C : scls[bidx];
    out[(size_t)B * L * 5 + base] = (float)label;   // labels (int -> f32 value)
    out[(size_t)B * L * 6 + base] = (float)bidx;    // target_inds (int -> f32 value)
}

extern "C" void kernel_launch(void* const* d_in, const int* in_sizes, int n_in,
                              void* d_out, int out_size, void* d_ws, size_t ws_size,
                              hipStream_t stream) {
    (void)n_in; (void)d_ws; (void)ws_size;
    const float* locations      = (const float*)d_in[0];
    const float* size_ranges    = (const float*)d_in[1];
    const float* stride_per_loc = (const float*)d_in[2];
    const float* gt_boxes       = (const float*)d_in[3];
    const int*   gt_classes     = (const int*)d_in[4];

    const int L  = in_sizes[2];              // #locations
    const int BG = in_sizes[4];              // B*G
    const int B  = out_size / (L * 7);       // outputs: B*L*5 + B*L + B*L
    int G = BG / (B > 0 ? B : 1);
    if (G > MAXG) G = MAXG;

    dim3 grid((L + BLOCK - 1) / BLOCK, B);
    fcos_assign_kernel<<<grid, BLOCK, 0, stream>>>(
        locations, size_ranges, stride_per_loc, gt_boxes, gt_classes,
        (float*)d_out, L, G, B);
}